// Feat_sampler_74775380623841
// MI455X (gfx1250) — compile-verified
//
#include <hip/hip_runtime.h>
#include <hip/hip_bf16.h>
#include <math.h>

// ---------------------------------------------------------------------------
// MI455X (gfx1250) implementation. All GEMM-shaped work goes through
// V_WMMA_F32_16X16X4_F32 (fp32 matrix path, preserves reference numerics).
// Problem constants (from reference): B=2, HC=256, H=W=128, INP_DIM=64.
// All hot-loop indexing is 32-bit; batch / row coordinates are block-uniform.
// ---------------------------------------------------------------------------

typedef __attribute__((ext_vector_type(2))) float v2f;
typedef __attribute__((ext_vector_type(8))) float v8f;

#define HWSZ   16384      // 128*128
#define NTOT   32768      // B*HWSZ
#define LDA_P  17         // 16 + 1 pad
#define LDB_P  33         // 32 + 1 pad

// WMMA fragment mapping (ISA 7.12.2):
//  A 16x4 : lane L holds row m=L&15, a[r] = A[m][ (L>>4)*2 + r ]
//  B 4x16 : lane L holds col n=L&15, b[r] = B[ (L>>4)*2 + r ][n]
//  C 16x16: lane L holds col n=L&15, c[r] = C[ (L>>4)*8 + r ][n]

// ---------------------------------------------------------------------------
// Generic 1x1-conv GEMM:  Y = act( W[M,K] * X + bias ), X/Y in NCHW (B,C,HW).
// block = 256 thr = 8 waves (4x2), tile = 64(M) x 32(N).
// ---------------------------------------------------------------------------
__global__ __launch_bounds__(256)
void k_gemm1x1(const float* __restrict__ Wt, const float* __restrict__ X,
               const float* __restrict__ bias, float* __restrict__ Y,
               int M, int K, int OC, int relu)
{
    __shared__ float lA[64 * LDA_P];
    __shared__ float lB[16 * LDB_P];
    const int tid  = threadIdx.x;
    const int lane = tid & 31;
    const int wave = tid >> 5;
    const int wm   = wave >> 1;     // 0..3
    const int wn   = wave & 1;      // 0..1
    const int half = lane >> 4;
    const int l16  = lane & 15;
    const int n_base = blockIdx.x * 32;
    const int m_base = blockIdx.y * 64;
    // 32-px tile lies within one batch: block-uniform b / hw0
    const int b   = n_base >> 14;
    const int hw0 = n_base & (HWSZ - 1);
    const float* __restrict__ Xb = X + b * (K * HWSZ) + hw0;

    for (int e = tid; e < 64 * LDA_P; e += 256) lA[e] = 0.f;  // pad rows stay 0

    v8f acc = {};
    for (int kc = 0; kc < K; kc += 16) {
        const int mrows = (M - m_base) < 64 ? (M - m_base) : 64;
        for (int e = tid; e < mrows * 16; e += 256) {
            int m = e >> 4, kk = e & 15;
            lA[m * LDA_P + kk] = Wt[(m_base + m) * K + kc + kk];
        }
        for (int e = tid; e < 16 * 32; e += 256) {
            int kk = e >> 5, n = e & 31;
            lB[kk * LDB_P + n] = Xb[(kc + kk) * HWSZ + n];
            if (kc + 16 < K)
                __builtin_prefetch(Xb + (kc + 16 + kk) * HWSZ + n, 0, 1);
        }
        __syncthreads();
        const float* lAr = lA + (wm * 16 + l16) * LDA_P;
        const float* lBc = lB + wn * 16 + l16;
        #pragma unroll
        for (int ks = 0; ks < 4; ++ks) {
            int kidx = ks * 4 + half * 2;
            v2f a, bb;
            a.x  = lAr[kidx];
            a.y  = lAr[kidx + 1];
            bb.x = lBc[kidx * LDB_P];
            bb.y = lBc[(kidx + 1) * LDB_P];
            acc = __builtin_amdgcn_wmma_f32_16x16x4_f32(
                false, a, false, bb, (short)0, acc, false, false);
        }
        __syncthreads();
    }
    const int hw = hw0 + wn * 16 + l16;
    #pragma unroll
    for (int r = 0; r < 8; ++r) {
        int m = m_base + wm * 16 + half * 8 + r;
        if (m < M) {
            float v = acc[r];
            if (bias) v += bias[m];
            if (relu) v = fmaxf(v, 0.f);
            Y[(b * OC + m) * HWSZ + hw] = v;
        }
    }
}

// ---------------------------------------------------------------------------
// 3x3 pad=1 conv as implicit GEMM (9 shifted GEMM passes).
// Input channels [0,C1) from X1, [C1,IC) from X2 (concat in sq conv).
// Weight layout (M, IC, 3, 3) row-major. Row shift is block-uniform.
// ---------------------------------------------------------------------------
__global__ __launch_bounds__(256)
void k_conv3x3(const float* __restrict__ Wt,
               const float* __restrict__ X1, int C1,
               const float* __restrict__ X2,
               const float* __restrict__ bias, float* __restrict__ Y,
               int M, int IC, int OC, int relu)
{
    __shared__ float lA[64 * LDA_P];
    __shared__ float lB[16 * LDB_P];
    const int tid  = threadIdx.x;
    const int lane = tid & 31;
    const int wave = tid >> 5;
    const int wm   = wave >> 1;
    const int wn   = wave & 1;
    const int half = lane >> 4;
    const int l16  = lane & 15;
    const int n_base = blockIdx.x * 32;     // 32 px within one image row
    const int b   = n_base >> 14;
    const int hw0 = n_base & (HWSZ - 1);
    const int x0  = hw0 & 127;
    const int y   = hw0 >> 7;
    const int C2  = IC - C1;

    for (int e = tid; e < 64 * LDA_P; e += 256) lA[e] = 0.f;

    v8f acc = {};
    for (int k = 0; k < 9; ++k) {
        const int dy = k / 3 - 1, dx = k % 3 - 1;
        const int yy = y + dy;                       // block-uniform
        const bool rowok = (unsigned)yy < 128u;
        for (int kc = 0; kc < IC; kc += 16) {
            for (int e = tid; e < M * 16; e += 256) {
                int m = e >> 4, kk = e & 15;
                lA[m * LDA_P + kk] = Wt[(m * IC + (kc + kk)) * 9 + k];
            }
            // chunk is entirely on one side of the concat (C1 % 16 == 0)
            const float* __restrict__ Xrow = (kc < C1)
                ? X1 + (b * C1 + kc) * HWSZ + yy * 128
                : X2 + (b * C2 + (kc - C1)) * HWSZ + yy * 128;
            for (int e = tid; e < 16 * 32; e += 256) {
                int kk = e >> 5, n = e & 31;
                int xx = x0 + n + dx;
                float v = 0.f;
                if (rowok && (unsigned)xx < 128u) {
                    v = Xrow[kk * HWSZ + xx];
                    if (kc + 16 < IC && kc + 16 != C1)
                        __builtin_prefetch(Xrow + (16 + kk) * HWSZ + xx, 0, 1);
                }
                lB[kk * LDB_P + n] = v;
            }
            __syncthreads();
            const float* lAr = lA + (wm * 16 + l16) * LDA_P;
            const float* lBc = lB + wn * 16 + l16;
            #pragma unroll
            for (int ks = 0; ks < 4; ++ks) {
                int kidx = ks * 4 + half * 2;
                v2f a, bb;
                a.x  = lAr[kidx];
                a.y  = lAr[kidx + 1];
                bb.x = lBc[kidx * LDB_P];
                bb.y = lBc[(kidx + 1) * LDB_P];
                acc = __builtin_amdgcn_wmma_f32_16x16x4_f32(
                    false, a, false, bb, (short)0, acc, false, false);
            }
            __syncthreads();
        }
    }
    const int hw = hw0 + wn * 16 + l16;
    #pragma unroll
    for (int r = 0; r < 8; ++r) {
        int m = wm * 16 + half * 8 + r;
        if (m < M) {
            float v = acc[r];
            if (bias) v += bias[m];
            if (relu) v = fmaxf(v, 0.f);
            Y[(b * OC + m) * HWSZ + hw] = v;
        }
    }
}

// ---------------------------------------------------------------------------
// Fused DCNv2: bilinear sample (zero-border * sigmoid(mask)) + 64x576 GEMM.
// F: (B,64,HW), OM: (B,27,HW), DW: (64,64,3,3), DB: (64,), G: (B,64,HW).
// ---------------------------------------------------------------------------
__global__ __launch_bounds__(256)
void k_dcn(const float* __restrict__ F, const float* __restrict__ OM,
           const float* __restrict__ DW, const float* __restrict__ DB,
           float* __restrict__ G)
{
    __shared__ float sA[64 * 65];    // tap weights (64x64, pad)
    __shared__ float sB[64 * LDB_P]; // sampled tile (64ic x 32px, pad)
    __shared__ int   sIdx[32][4];
    __shared__ float sWt[32][4];
    const int tid  = threadIdx.x;
    const int lane = tid & 31;
    const int wave = tid >> 5;
    const int wm   = wave >> 1;
    const int wn   = wave & 1;
    const int half = lane >> 4;
    const int l16  = lane & 15;
    const int n_base = blockIdx.x * 32;
    const int b   = n_base >> 14;                 // block-uniform batch
    const int hw0 = n_base & (HWSZ - 1);
    const float* __restrict__ Fb  = F + b * (64 * HWSZ);
    const float* __restrict__ omb = OM + b * (27 * HWSZ) + hw0;

    v8f acc = {};
    for (int k = 0; k < 9; ++k) {
        if (tid < 32) {
            int hw = hw0 + tid;
            int x = hw & 127, yv = hw >> 7;
            float dyv = omb[(2 * k) * HWSZ + tid];      // offset ch 0::2 = dy
            float dxv = omb[(2 * k + 1) * HWSZ + tid];  // offset ch 1::2 = dx
            float mk  = 1.f / (1.f + __expf(-omb[(18 + k) * HWSZ + tid]));
            float py = (float)yv + (float)(k / 3 - 1) + dyv;
            float px = (float)x  + (float)(k % 3 - 1) + dxv;
            float fx0 = floorf(px), fy0 = floorf(py);
            float wx1 = px - fx0, wx0 = 1.f - wx1;
            float wy1 = py - fy0, wy0 = 1.f - wy1;
            float cw[4]  = {wx0 * wy0, wx1 * wy0, wx0 * wy1, wx1 * wy1};
            float cxs[4] = {fx0, fx0 + 1.f, fx0, fx0 + 1.f};
            float cys[4] = {fy0, fy0, fy0 + 1.f, fy0 + 1.f};
            #pragma unroll
            for (int j = 0; j < 4; ++j) {
                float cx = cxs[j], cy = cys[j];
                bool valid = (cx >= 0.f) && (cx <= 127.f) &&
                             (cy >= 0.f) && (cy <= 127.f);
                int xi = (int)fminf(fmaxf(cx, 0.f), 127.f);
                int yi = (int)fminf(fmaxf(cy, 0.f), 127.f);
                sIdx[tid][j] = yi * 128 + xi;
                sWt[tid][j]  = valid ? cw[j] * mk : 0.f;
            }
        }
        __syncthreads();
        for (int e = tid; e < 64 * 32; e += 256) {
            int ic = e >> 5, p = e & 31;
            const float* fp = Fb + ic * HWSZ;
            sB[ic * LDB_P + p] =
                  sWt[p][0] * fp[sIdx[p][0]] + sWt[p][1] * fp[sIdx[p][1]]
                + sWt[p][2] * fp[sIdx[p][2]] + sWt[p][3] * fp[sIdx[p][3]];
        }
        for (int e = tid; e < 64 * 64; e += 256) {
            int o = e >> 6, ic = e & 63;
            sA[o * 65 + ic] = DW[(o * 64 + ic) * 9 + k];
        }
        __syncthreads();
        const float* sAr = sA + (wm * 16 + l16) * 65;
        const float* sBc = sB + wn * 16 + l16;
        #pragma unroll
        for (int ks = 0; ks < 16; ++ks) {
            int kidx = ks * 4 + half * 2;
            v2f a, bb;
            a.x  = sAr[kidx];
            a.y  = sAr[kidx + 1];
            bb.x = sBc[kidx * LDB_P];
            bb.y = sBc[(kidx + 1) * LDB_P];
            acc = __builtin_amdgcn_wmma_f32_16x16x4_f32(
                false, a, false, bb, (short)0, acc, false, false);
        }
        __syncthreads();
    }
    const int hw = hw0 + wn * 16 + l16;
    #pragma unroll
    for (int r = 0; r < 8; ++r) {
        int m = wm * 16 + half * 8 + r;
        G[(b * 64 + m) * HWSZ + hw] = acc[r] + DB[m];
    }
}

// ---------------------------------------------------------------------------
// BatchNorm stats: one block per channel (64 ch), reduce over B*HW = 32768.
// ---------------------------------------------------------------------------
__global__ __launch_bounds__(256)
void k_bnstats(const float* __restrict__ X, float* __restrict__ MEAN,
               float* __restrict__ INV)
{
    __shared__ float rs[256], rs2[256];
    const int c = blockIdx.x, tid = threadIdx.x;
    float s = 0.f, s2 = 0.f;
    for (int i = tid; i < NTOT; i += 256) {
        int b = i >> 14, hw = i & (HWSZ - 1);
        float v = X[(b * 64 + c) * HWSZ + hw];
        s += v; s2 += v * v;
    }
    rs[tid] = s; rs2[tid] = s2;
    __syncthreads();
    for (int o = 128; o > 0; o >>= 1) {
        if (tid < o) { rs[tid] += rs[tid + o]; rs2[tid] += rs2[tid + o]; }
        __syncthreads();
    }
    if (tid == 0) {
        float m = rs[0] / 32768.f;
        float var = rs2[0] / 32768.f - m * m;
        MEAN[c] = m;
        INV[c]  = rsqrtf(var + 1e-5f);
    }
}

__global__ __launch_bounds__(256)
void k_bnrelu(float* __restrict__ X, const float* __restrict__ MEAN,
              const float* __restrict__ INV, const float* __restrict__ Gm,
              const float* __restrict__ Bt)
{
    int i = blockIdx.x * 256 + threadIdx.x;   // total 2*64*HW = 2097152
    int c = (i >> 14) & 63;
    float v = Gm[c] * (X[i] - MEAN[c]) * INV[c] + Bt[c];
    X[i] = fmaxf(v, 0.f);
}

// ---------------------------------------------------------------------------
// Grid-resample (border=True bilinear) of a (B,64,HW) tensor by adapt offsets
// (= offset1 channels [2h, 2h+1]). px = (offx+x)*W/(W-1) - 0.5, clipped.
// ---------------------------------------------------------------------------
__global__ __launch_bounds__(256)
void k_resample(const float* __restrict__ SRC, const float* __restrict__ OFF1,
                int head, float* __restrict__ DST, int dstC, int cbase)
{
    __shared__ int   sIdx[64][4];
    __shared__ float sWt[64][4];
    const int tid = threadIdx.x;
    const int p0  = blockIdx.x * 64;
    const int b   = p0 >> 14;                  // block-uniform batch
    const int hw0 = p0 & (HWSZ - 1);
    if (tid < 64) {
        int hw = hw0 + tid;
        int x = hw & 127, y = hw >> 7;
        const float* ab = OFF1 + (b * 14 + 2 * head) * HWSZ;
        float offx = ab[hw];
        float offy = ab[HWSZ + hw];
        const float sc = 128.f / 127.f;
        float px = (offx + (float)x) * sc - 0.5f;
        float py = (offy + (float)y) * sc - 0.5f;
        px = fminf(fmaxf(px, 0.f), 127.f);
        py = fminf(fmaxf(py, 0.f), 127.f);
        float fx0 = floorf(px), fy0 = floorf(py);
        float wx1 = px - fx0, wx0 = 1.f - wx1;
        float wy1 = py - fy0, wy0 = 1.f - wy1;
        float cw[4]  = {wx0 * wy0, wx1 * wy0, wx0 * wy1, wx1 * wy1};
        float cxs[4] = {fx0, fx0 + 1.f, fx0, fx0 + 1.f};
        float cys[4] = {fy0, fy0, fy0 + 1.f, fy0 + 1.f};
        #pragma unroll
        for (int j = 0; j < 4; ++j) {
            int xi = (int)fminf(fmaxf(cxs[j], 0.f), 127.f);
            int yi = (int)fminf(fmaxf(cys[j], 0.f), 127.f);
            sIdx[tid][j] = yi * 128 + xi;
            sWt[tid][j]  = cw[j];            // border=True: no validity mask
        }
    }
    __syncthreads();
    const int p  = tid & 63;
    const int c0 = tid >> 6;
    const int hw = hw0 + p;
    for (int c = c0; c < 64; c += 4) {
        const float* sp = SRC + (b * 64 + c) * HWSZ;
        float v = sWt[p][0] * sp[sIdx[p][0]] + sWt[p][1] * sp[sIdx[p][1]]
                + sWt[p][2] * sp[sIdx[p][2]] + sWt[p][3] * sp[sIdx[p][3]];
        DST[(b * dstC + cbase + c) * HWSZ + hw] = v;
    }
}

// ---------------------------------------------------------------------------
// Per-head pred 1x1 conv (<=10 out ch) + adapt add + scatter into kps layout.
// ---------------------------------------------------------------------------
struct CMap { int c[10]; };

__global__ __launch_bounds__(256)
void k_pred(const float* __restrict__ KF, const float* __restrict__ PW,
            const float* __restrict__ PB, const float* __restrict__ OFF1,
            int head, int nOut, CMap map, float* __restrict__ KPS)
{
    const int gn  = blockIdx.x * 256 + threadIdx.x;
    const int b   = gn >> 14;
    const int hw  = gn & (HWSZ - 1);
    float reg[64];
    const float* kf = KF + b * (64 * HWSZ) + hw;
    #pragma unroll
    for (int ic = 0; ic < 64; ++ic) reg[ic] = kf[ic * HWSZ];
    const float* ab = OFF1 + (b * 14 + 2 * head) * HWSZ;
    float a0 = ab[hw], a1 = ab[HWSZ + hw];
    for (int j = 0; j < nOut; ++j) {
        float acc = PB[j];
        #pragma unroll
        for (int ic = 0; ic < 64; ++ic) acc += PW[j * 64 + ic] * reg[ic];
        acc += (j & 1) ? a1 : a0;
        KPS[(b * 34 + map.c[j]) * HWSZ + hw] = acc;
    }
}

// hm 1x1 conv 64 -> 1
__global__ __launch_bounds__(256)
void k_hm(const float* __restrict__ CTHM, const float* __restrict__ HWt,
          const float* __restrict__ HB, float* __restrict__ CT)
{
    const int gn = blockIdx.x * 256 + threadIdx.x;
    const int b  = gn >> 14;
    const int hw = gn & (HWSZ - 1);
    float acc = HB[0];
    #pragma unroll
    for (int ic = 0; ic < 64; ++ic)
        acc += HWt[ic] * CTHM[(b * 64 + ic) * HWSZ + hw];
    CT[gn] = acc;
}

// ---------------------------------------------------------------------------
extern "C" void kernel_launch(void* const* d_in, const int* in_sizes, int n_in,
                              void* d_out, int out_size, void* d_ws, size_t ws_size,
                              hipStream_t stream)
{
    (void)in_sizes; (void)n_in; (void)out_size; (void)ws_size;
    // setup_inputs() insertion order, params dict expanded in order, lists per-element:
    const float* kps_feat = (const float*)d_in[0];
    const float* ct_feat  = (const float*)d_in[1];
    const float* searcher_w = (const float*)d_in[2];
    const float* searcher_b = (const float*)d_in[3];
    const float *trans_w[7], *bn_g[7], *bn_b[7], *om_w[7], *om_b[7];
    const float *dcn_w[7], *dcn_b[7], *pred_w[7], *pred_b[7];
    for (int i = 0; i < 7; ++i) {
        trans_w[i] = (const float*)d_in[4 + i];
        bn_g[i]    = (const float*)d_in[11 + i];
        bn_b[i]    = (const float*)d_in[18 + i];
        om_w[i]    = (const float*)d_in[25 + i];
        om_b[i]    = (const float*)d_in[32 + i];
        dcn_w[i]   = (const float*)d_in[39 + i];
        dcn_b[i]   = (const float*)d_in[46 + i];
        pred_w[i]  = (const float*)d_in[53 + i];
        pred_b[i]  = (const float*)d_in[60 + i];
    }
    const float* ct_w = (const float*)d_in[67];
    const float* ct_b = (const float*)d_in[68];
    const float* sq_w = (const float*)d_in[69];
    const float* sq_b = (const float*)d_in[70];
    const float* hm_w = (const float*)d_in[71];
    const float* hm_b = (const float*)d_in[72];

    float* out  = (float*)d_out;
    float* kps  = out;                 // (2,34,128,128)
    float* ct   = out + 1114112;       // (2, 1,128,128)
    float* off1 = out + 1146880;       // (2,14,128,128)

    // workspace carve (~92 MB fp32)
    float* ws      = (float*)d_ws;
    float* ws_cttr = ws;  ws += (size_t)2 * 64  * HWSZ;   // ct_tr
    float* ws_cf   = ws;  ws += (size_t)2 * 448 * HWSZ;   // cf for all 7 heads
    float* ws_f    = ws;  ws += (size_t)2 * 64  * HWSZ;   // trans/bn out, reused as kf
    float* ws_om   = ws;  ws += (size_t)2 * 27  * HWSZ;   // dcn offset/mask
    float* ws_g    = ws;  ws += (size_t)2 * 64  * HWSZ;   // dcn out
    float* ws_cthm = ws;  ws += (size_t)2 * 64  * HWSZ;   // sq conv out
    float* ws_mean = ws;  ws += 64;
    float* ws_inv  = ws;  ws += 64;

    static const int head_n[7] = {5, 2, 2, 2, 2, 2, 2};
    static const int cmap_tab[7][10] = {
        {0, 1, 2, 3, 4, 5, 6, 7, 8, 9},          // face
        {10, 11, 12, 13, 0, 0, 0, 0, 0, 0},      // sh
        {14, 15, 18, 19, 0, 0, 0, 0, 0, 0},      // lew -> [:2],[2:]
        {16, 17, 20, 21, 0, 0, 0, 0, 0, 0},      // rew
        {22, 23, 24, 25, 0, 0, 0, 0, 0, 0},      // hip
        {26, 27, 30, 31, 0, 0, 0, 0, 0, 0},      // lka
        {28, 29, 32, 33, 0, 0, 0, 0, 0, 0}       // rka
    };

    dim3 blk(256);
    // offset1 = searcher 3x3 conv (256->14), written directly into d_out
    k_conv3x3<<<dim3(1024, 1), blk, 0, stream>>>(searcher_w, kps_feat, 256,
                                                 nullptr, searcher_b, off1,
                                                 14, 256, 14, 0);
    // ct_tr = 1x1 conv 256->64
    k_gemm1x1<<<dim3(1024, 1), blk, 0, stream>>>(ct_w, ct_feat, ct_b, ws_cttr,
                                                 64, 256, 64, 0);
    for (int i = 0; i < 7; ++i) {
        // trans 1x1 (no bias), then BN stats + bn/relu in place
        k_gemm1x1<<<dim3(1024, 1), blk, 0, stream>>>(trans_w[i], kps_feat,
                                                     nullptr, ws_f, 64, 256, 64, 0);
        k_bnstats<<<64, blk, 0, stream>>>(ws_f, ws_mean, ws_inv);
        k_bnrelu<<<8192, blk, 0, stream>>>(ws_f, ws_mean, ws_inv, bn_g[i], bn_b[i]);
        // dcn: om 3x3 conv (64->27), then fused sample+GEMM
        k_conv3x3<<<dim3(1024, 1), blk, 0, stream>>>(om_w[i], ws_f, 64, nullptr,
                                                     om_b[i], ws_om, 27, 64, 27, 0);
        k_dcn<<<1024, blk, 0, stream>>>(ws_f, ws_om, dcn_w[i], dcn_b[i], ws_g);
        // kf = resample(g, adapt) (reuse ws_f), cf = resample(ct_tr, adapt)
        k_resample<<<512, blk, 0, stream>>>(ws_g, off1, i, ws_f, 64, 0);
        k_resample<<<512, blk, 0, stream>>>(ws_cttr, off1, i, ws_cf, 448, i * 64);
        // off2 + tile(adapt) scattered into kps channels
        CMap m;
        for (int j = 0; j < 10; ++j) m.c[j] = cmap_tab[i][j];
        k_pred<<<128, blk, 0, stream>>>(ws_f, pred_w[i], pred_b[i], off1, i,
                                        head_n[i] * 2, m, kps);
    }
    // ct_hm = relu(sq 3x3 conv over concat(ct_feat[256], cf_all[448]))
    k_conv3x3<<<dim3(1024, 1), blk, 0, stream>>>(sq_w, ct_feat, 256, ws_cf,
                                                 sq_b, ws_cthm, 64, 704, 64, 1);
    // ct = hm 1x1 conv 64->1
    k_hm<<<128, blk, 0, stream>>>(ws_cthm, hm_w, hm_b, ct);
}